// GeoLLMLayer_11381663334993
// MI455X (gfx1250) — compile-verified
//
#include <hip/hip_runtime.h>
#include <hip/hip_bf16.h>

typedef float v2f __attribute__((ext_vector_type(2)));
typedef float v8f __attribute__((ext_vector_type(8)));

#define B_  2
#define S_  8192
#define D_  4096
#define BS_ (B_*S_)
#define CH  128   // D-chunk staged in LDS per iteration

// ---------------------------------------------------------------------------
// Kernel 1: H[row][k] = sum_d x[row][d] * w_in[k][d] + b_in[k]   (row in B*S)
// WMMA f32 16x16x4: A = x tile (16 rows x 4 D-cols), B = w_in^T (4 x 16, cols
// >=3 zero-padded), C accumulates over all of D.  A/B fragments per ISA
// 7.12.2: lane L holds (M=L&15, K = 2*half .. 2*half+1) => ds_load_b64.
// ---------------------------------------------------------------------------
__global__ __launch_bounds__(128) void proj_in_wmma(const float* __restrict__ x,
                                                    const float* __restrict__ w_in,
                                                    const float* __restrict__ b_in,
                                                    float* __restrict__ H) {
    __shared__ float wlds[16 * CH];        // w_in^T chunk, rows 3..15 zero
    __shared__ float xlds[4 * 16 * CH];    // per-wave 16 x CH x tile

    const int tid   = threadIdx.x;
    const int wv    = tid >> 5;
    const int lane  = tid & 31;
    const int m     = lane & 15;           // A row / B col / C col index
    const int half  = lane >> 4;
    const int half2 = half << 1;           // K offset: 0 or 2
    const int row0  = blockIdx.x * 64 + wv * 16;

    // zero-pad B rows 3..15 once (they are never rewritten)
    for (int i = tid; i < 13 * CH; i += 128) wlds[3 * CH + i] = 0.0f;

    v8f c = {};
    for (int c0 = 0; c0 < D_; c0 += CH) {
        __syncthreads();                   // previous compute done (and pad done)
        // stage w_in rows 0..2 of this chunk (blockDim == CH)
        for (int r = 0; r < 3; ++r)
            wlds[r * CH + tid] = w_in[(size_t)r * D_ + c0 + tid];
        // stage this wave's 16 x CH x-tile, coalesced float4 per lane
        {
            float* xw = &xlds[wv * 16 * CH];
            for (int it = 0; it < 16; ++it) {
                const float4 v = *(const float4*)&x[(size_t)(row0 + it) * D_ + c0 + lane * 4];
                *(float4*)&xw[it * CH + lane * 4] = v;
            }
        }
        __syncthreads();

        const float* xa = &xlds[wv * 16 * CH + m * CH + half2]; // A row m
        const float* wb = &wlds[m * CH + half2];                // B col n (= m)
        #pragma unroll
        for (int d0 = 0; d0 < CH; d0 += 4) {
            v2f a = *(const v2f*)(xa + d0);
            v2f b = *(const v2f*)(wb + d0);
            c = __builtin_amdgcn_wmma_f32_16x16x4_f32(false, a, false, b,
                                                      (short)0, c, false, false);
        }
    }

    // C layout: VGPR j -> (M = j + half*8, N = lane&15); keep N < 3
    if (m < 3) {
        const float bias  = b_in[m];
        const int   rbase = row0 + (half ? 8 : 0);
        #pragma unroll
        for (int j = 0; j < 8; ++j)
            H[(size_t)(rbase + j) * 3 + m] = c[j] + bias;
    }
}

// ---------------------------------------------------------------------------
// Kernel 2: per-channel (b,k) inclusive scan of h*cos / h*sin, then
// out_geom = re*cos + im*sin.  THETA derived on-device in double from the
// tribonacci cubic (Newton) -> no hand-rounded constant.
// ---------------------------------------------------------------------------
__global__ __launch_bounds__(256) void geom_scan(const float* __restrict__ H,
                                                 float* __restrict__ OG) {
    __shared__ float sre[256], sim[256];
    const int ch  = blockIdx.x;            // 0..5
    const int b   = ch / 3, k = ch % 3;
    const int tid = threadIdx.x;

    double t = 1.8392867552141612;         // tribonacci constant seed
    #pragma unroll
    for (int i = 0; i < 3; ++i) {          // Newton on t^3 - t^2 - t - 1
        double f  = ((t - 1.0) * t - 1.0) * t - 1.0;
        double fp = (3.0 * t - 2.0) * t - 1.0;
        t -= f / fp;
    }
    const double rel = (1.0 - t) * 0.5;                 // Re(lam2)
    const double iml = sqrt(1.0 / t - rel * rel);       // Im(lam2), |lam2|^2 = 1/t
    const float  th  = (float)atan2(iml, rel);          // float32, as reference

    const int base = tid * 32;                          // 256 thr x 32 = 8192
    const float* Hc = H + (size_t)b * S_ * 3 + k;

    float sc = 0.f, ss = 0.f;
    for (int j = 0; j < 32; ++j) {
        const int tt = base + j;
        double sd, cd;
        sincos((double)th * (double)tt, &sd, &cd);
        const float h = Hc[(size_t)tt * 3];
        sc += h * (float)cd;
        ss += h * (float)sd;
    }
    sre[tid] = sc; sim[tid] = ss;
    __syncthreads();
    if (tid == 0) {                                     // tiny serial prefix
        float r = 0.f, i = 0.f;
        for (int q = 0; q < 256; ++q) {
            const float vr = sre[q], vi = sim[q];
            sre[q] = r; sim[q] = i;
            r += vr; i += vi;
        }
    }
    __syncthreads();

    float re = sre[tid], im = sim[tid];
    float* OGc = OG + (size_t)b * S_ * 3 + k;
    for (int j = 0; j < 32; ++j) {
        const int tt = base + j;
        double sd, cd;
        sincos((double)th * (double)tt, &sd, &cd);
        const float cf = (float)cd, sf = (float)sd;
        const float h  = Hc[(size_t)tt * 3];
        re += h * cf;
        im += h * sf;
        OGc[(size_t)tt * 3] = re * cf + im * sf;
    }
}

// ---------------------------------------------------------------------------
// Kernel 3: out[row][d] = sum_k out_geom[row][k] * w_out[d][k] + b_out[d]
// WMMA f32 16x16x4: A = out_geom tile (16 x 4, K=3 padded), B = w_out^T tile
// (4 x 16).  Bias injected via C (all 8 C regs = b_out[d0+n], since C value
// depends only on N).  One wave: fixed row tile, 64 column tiles.
// ---------------------------------------------------------------------------
__global__ __launch_bounds__(256) void proj_out_wmma(const float* __restrict__ OG,
                                                     const float* __restrict__ w_out,
                                                     const float* __restrict__ b_out,
                                                     float* __restrict__ out) {
    const int tid  = threadIdx.x;
    const int wv   = tid >> 5;
    const int lane = tid & 31;
    const int half = lane >> 4;
    const int n    = lane & 15;            // A row m == C col n per lane
    const int rt   = blockIdx.x * 8 + wv;  // row tile (0..1023)

    // A fragment: lane holds (M=n, K=2*half .. 2*half+1); K=3 is zero pad
    const size_t arow = (size_t)(rt * 16 + n) * 3;
    v2f a;
    a.x = half ? OG[arow + 2] : OG[arow + 0];
    a.y = half ? 0.0f         : OG[arow + 1];

    const int    ct0     = blockIdx.y * 64;
    const size_t rowbase = (size_t)(rt * 16 + half * 8);

    for (int ct = ct0; ct < ct0 + 64; ++ct) {
        const int d0 = ct * 16;
        const int d  = d0 + n;
        const float w0 = w_out[(size_t)d * 3 + 0];
        const float w1 = w_out[(size_t)d * 3 + 1];
        const float w2 = w_out[(size_t)d * 3 + 2];
        v2f b;                              // B[k][n] = w_out[d0+n][k]
        b.x = half ? w2 : w0;               // K = 2*half
        b.y = half ? 0.0f : w1;             // K = 2*half+1 (K=3 pad)
        const float bias = b_out[d];
        v8f c = {bias, bias, bias, bias, bias, bias, bias, bias};
        c = __builtin_amdgcn_wmma_f32_16x16x4_f32(false, a, false, b,
                                                  (short)0, c, false, false);
        #pragma unroll
        for (int j = 0; j < 8; ++j)         // VGPR j -> row (half*8 + j), col n
            out[(rowbase + j) * (size_t)D_ + d] = c[j];
    }
}

// ---------------------------------------------------------------------------
extern "C" void kernel_launch(void* const* d_in, const int* in_sizes, int n_in,
                              void* d_out, int out_size, void* d_ws, size_t ws_size,
                              hipStream_t stream) {
    const float* x     = (const float*)d_in[0];
    const float* w_in  = (const float*)d_in[1];
    const float* b_in  = (const float*)d_in[2];
    const float* w_out = (const float*)d_in[3];
    const float* b_out = (const float*)d_in[4];
    float*       out   = (float*)d_out;

    float* H  = (float*)d_ws;              // (B*S, 3)
    float* OG = H + (size_t)BS_ * 3;       // (B*S, 3); total ws use ~384 KB

    proj_in_wmma <<<dim3(BS_ / 64), dim3(128), 0, stream>>>(x, w_in, b_in, H);
    geom_scan    <<<dim3(6),        dim3(256), 0, stream>>>(H, OG);
    proj_out_wmma<<<dim3(BS_ / 128, 4), dim3(256), 0, stream>>>(OG, w_out, b_out, out);
}